// BiquadCell_3332894622689
// MI455X (gfx1250) — compile-verified
//
#include <hip/hip_runtime.h>
#include <stdint.h>

// Biquad IIR scan, B=256, T=65536, F=3 taps — CDNA5 (gfx1250).
// Three-pass linear-recurrence (scan) decomposition:
//   pass 1: zero-state response per (batch-group, chunk) strip.
//           Per 16x16 tile:  Y = H * F  via 4x V_WMMA_F32_16X16X4_F32,
//           x staged global->LDS with async b128 copies (ASYNCcnt,
//           double-buffered), state carried with 2 FMAs + ds_bpermute.
//   pass 2: tiny 2x2 affine scan over chunk boundaries.
//   pass 3: superposition fixup, truncated where h underflows to exact 0.
// Roofline: ~300 MB HBM traffic @ 23.3 TB/s ~= 13 us; compute negligible.

typedef __attribute__((ext_vector_type(2))) float v2f;
typedef __attribute__((ext_vector_type(4))) float v4f;
typedef __attribute__((ext_vector_type(8))) float v8f;
typedef __attribute__((ext_vector_type(4))) int   v4i;

#define B_    256
#define T_    65536
#define C_    256              // chunks per batch (time split)
#define LC    (T_ / C_)        // 256 timesteps per chunk
#define NTILE (LC / 16)        // 16 WMMA tiles per strip
#define BG_   (B_ / 16)        // 16 batch-groups of 16
#define WPB   8                // waves per block

#define ROW_DW 52              // LDS row stride in dwords (48 data + 4 pad)

// ---- gfx1250 async global->LDS copy + ASYNCcnt waits -----------------------
// Builtin wants int4* operands (per hipcc diagnostic).  Cast to AS1/AS3-
// qualified int4*: exact match if the params are AS-qualified, legal implicit
// superset conversion if they are generic.
typedef __attribute__((address_space(1))) v4i* gv4i_p;
typedef __attribute__((address_space(3))) v4i* lv4i_p;
typedef __attribute__((address_space(3))) void lvoid_t;

__device__ __forceinline__ void async_copy16(const float* gp, float* lp) {
#if __has_builtin(__builtin_amdgcn_global_load_async_to_lds_b128)
  __builtin_amdgcn_global_load_async_to_lds_b128(
      (gv4i_p)(void*)gp, (lv4i_p)lp, 0, 0);
#else
  const unsigned lds_off = (unsigned)(unsigned long long)(lvoid_t*)lp;
  asm volatile("global_load_async_to_lds_b128 %0, %1, off"
               :: "v"(lds_off), "v"(gp) : "memory");
#endif
}

__device__ __forceinline__ void wait_async_le6() {
#if __has_builtin(__builtin_amdgcn_s_wait_asynccnt)
  __builtin_amdgcn_s_wait_asynccnt(6);
#else
  asm volatile("s_wait_asynccnt 0x6" ::: "memory");
#endif
}
__device__ __forceinline__ void wait_async_le0() {
#if __has_builtin(__builtin_amdgcn_s_wait_asynccnt)
  __builtin_amdgcn_s_wait_asynccnt(0);
#else
  asm volatile("s_wait_asynccnt 0x0" ::: "memory");
#endif
}

// ---------------------------------------------------------------------------
// Pass 1: one wave per strip (16 batches x 256 timesteps); 4096 strips.
// ---------------------------------------------------------------------------
__global__ __launch_bounds__(256) void biquad_pass1(
    const float* __restrict__ x, const float* __restrict__ coef,
    float* __restrict__ y) {
  const int lane = threadIdx.x & 31;
  const int wib  = threadIdx.x >> 5;
  const int w    = blockIdx.x * WPB + wib;   // strip id
  const int c    = w % C_;                   // chunk index
  const int bg   = w / C_;                   // batch group
  const int hi   = lane >> 4;                // lane half
  const int n    = lane & 15;                // N col / A row

  const float b0 = coef[0], b1 = coef[1], b2 = coef[2];
  const float a1 = coef[3], a2 = coef[4];

  // Impulse response h[0..16] in LDS (identical across waves).
  __shared__ float hs[20];
  // Per-wave double-buffered x tile: 16 rows x 52 dwords (padded; the pad
  // makes 52*n mod 64 distinct over n=0..15 and the two lane halves land in
  // different mod-4 bank classes -> conflict-free ds reads).
  __shared__ float xs[WPB][2][16 * ROW_DW];

  if (threadIdx.x == 0) {
    hs[0] = 1.0f; hs[1] = a1;
    for (int k = 2; k <= 16; ++k) hs[k] = a1 * hs[k - 1] + a2 * hs[k - 2];
  }
  __syncthreads();

  // A-operand fragments: H[m][k] = (k<=m) ? h[m-k] : 0 ;  m = lane%16,
  // K = 4*blk + 2*hi + v  (ISA split-half layout for 32-bit 16x4 A).
  v2f afrag[4];
#pragma unroll
  for (int blk = 0; blk < 4; ++blk) {
#pragma unroll
    for (int v = 0; v < 2; ++v) {
      const int idx = n - (4 * blk + 2 * hi + v);
      afrag[blk][v] = (idx >= 0) ? hs[idx] : 0.0f;
    }
  }
  // State-injection coefficients for D rows t = r + 8*hi:
  // y[t] += h[t+1]*p1 + (a2*h[t])*p2
  float hp1[8], hq[8];
#pragma unroll
  for (int r = 0; r < 8; ++r) {
    hp1[r] = hs[r + 8 * hi + 1];
    hq[r]  = a2 * hs[r + 8 * hi];
  }

  const int  t0      = c * LC;
  const long rowBase = (long)(bg * 16 + n) * T_;  // this lane's batch row

  // Async-copy address plan: 192 16B chunks per tile; lane handles
  // idx = i*32+lane -> row r = idx/12, col = idx%12 (row = 192B contiguous).
  const float* gbase[6];
  float*       lbase[2][6];
#pragma unroll
  for (int i = 0; i < 6; ++i) {
    const int idx = i * 32 + lane;
    const int r   = idx / 12;
    const int col = idx - r * 12;
    gbase[i]    = x + ((long)(bg * 16 + r) * T_ + t0) * 3 + col * 4;
    lbase[0][i] = &xs[wib][0][0] + r * ROW_DW + col * 4;
    lbase[1][i] = &xs[wib][1][0] + r * ROW_DW + col * 4;
  }

  // Prime the pipeline: tile 0 into buffer 0.
#pragma unroll
  for (int i = 0; i < 6; ++i) async_copy16(gbase[i], lbase[0][i]);

  float p1 = 0.0f, p2 = 0.0f;  // zero initial state
  for (int k = 0; k < NTILE; ++k) {
    const int tt = t0 + k * 16;

    if (k + 1 < NTILE) {  // fetch tile k+1 while we compute tile k
#pragma unroll
      for (int i = 0; i < 6; ++i)
        async_copy16(gbase[i] + (long)(k + 1) * 48, lbase[(k + 1) & 1][i]);
      wait_async_le6();   // in-order completion: tile k's 6 copies done
    } else {
      wait_async_le0();
    }

    // B-operand from LDS: F[t][n] = b.x, t = 4*blk + 2*hi + v.
    const float* xw = &xs[wib][k & 1][0] + n * ROW_DW;
    v2f bfrag[4];
#pragma unroll
    for (int blk = 0; blk < 4; ++blk) {
#pragma unroll
      for (int v = 0; v < 2; ++v) {
        const int tl = 4 * blk + 2 * hi + v;
        const float x0 = xw[tl * 3 + 0];
        const float x1 = xw[tl * 3 + 1];
        const float x2 = xw[tl * 3 + 2];
        bfrag[blk][v] = b0 * x0 + b1 * x1 + b2 * x2;
      }
    }

    // Y = H * F : 4 K-chained f32 WMMAs (16x16x4 each).
    v8f acc = {0.f, 0.f, 0.f, 0.f, 0.f, 0.f, 0.f, 0.f};
#pragma unroll
    for (int blk = 0; blk < 4; ++blk) {
      acc = __builtin_amdgcn_wmma_f32_16x16x4_f32(
          false, afrag[blk], false, bfrag[blk], (short)0, acc, false, false);
    }

    // Inject carry-in state (the only serial dependency across tiles).
#pragma unroll
    for (int r = 0; r < 8; ++r) acc[r] += hp1[r] * p1 + hq[r] * p2;

    // New state = (y[15], y[14]) = (acc[7], acc[6]) of lanes 16..31.
    const int src = (n | 16) << 2;
    p1 = __int_as_float(__builtin_amdgcn_ds_bpermute(src, __float_as_int(acc[7])));
    p2 = __int_as_float(__builtin_amdgcn_ds_bpermute(src, __float_as_int(acc[6])));

    // Each lane holds t = r + 8*hi (r=0..7): two aligned b128 stores.
    v4f s0 = {acc[0], acc[1], acc[2], acc[3]};
    v4f s1 = {acc[4], acc[5], acc[6], acc[7]};
    *(v4f*)(y + rowBase + tt + 8 * hi)     = s0;
    *(v4f*)(y + rowBase + tt + 8 * hi + 4) = s1;
  }
}

// ---------------------------------------------------------------------------
// Pass 2: sequential scan over chunks (per batch): s_c = A^L s_{c-1} + v_{c-1}
// ---------------------------------------------------------------------------
__global__ __launch_bounds__(256) void biquad_pass2(
    const float* __restrict__ y0, const float* __restrict__ carry0,
    const float* __restrict__ coef, float* __restrict__ S) {
  const int b = threadIdx.x;  // 256 threads == 256 batches
  const float a1 = coef[3], a2 = coef[4];

  // M = A^LC, A = [[a1, a2], [1, 0]]  (decays to ~0 for stable poles)
  float m00 = 1.f, m01 = 0.f, m10 = 0.f, m11 = 1.f;
  for (int k = 0; k < LC; ++k) {
    const float t0 = a1 * m00 + a2 * m10;
    const float t1 = a1 * m01 + a2 * m11;
    m10 = m00; m11 = m01; m00 = t0; m01 = t1;
  }

  float s1 = carry0[2 * b + 0];  // y[-1]
  float s2 = carry0[2 * b + 1];  // y[-2]
  const float* yrow = y0 + (long)b * T_;
  for (int c = 0; c < C_; ++c) {
    S[((long)b * C_ + c) * 2 + 0] = s1;
    S[((long)b * C_ + c) * 2 + 1] = s2;
    const float vz1 = yrow[c * LC + LC - 1];
    const float vz2 = yrow[c * LC + LC - 2];
    const float ns1 = m00 * s1 + m01 * s2 + vz1;
    const float ns2 = m10 * s1 + m11 * s2 + vz2;
    s1 = ns1; s2 = ns2;
  }
}

// ---------------------------------------------------------------------------
// Pass 3: y[t] += h[t+1]*s1 + (a2*h[t])*s2 per chunk; truncated at the exact
// FP-underflow point of h (two consecutive 0.0f -> zero forever).
// ---------------------------------------------------------------------------
__global__ __launch_bounds__(256) void biquad_pass3(
    float* __restrict__ y, const float* __restrict__ S,
    const float* __restrict__ coef) {
  __shared__ float hs[LC + 2];
  __shared__ int tmax_s;
  const int bg = blockIdx.x / C_;
  const int c  = blockIdx.x % C_;
  const float a1 = coef[3], a2 = coef[4];

  if (threadIdx.x == 0) {
    hs[0] = 1.f; hs[1] = a1;
    for (int k = 2; k <= LC; ++k) hs[k] = a1 * hs[k - 1] + a2 * hs[k - 2];
    int tm = LC - 1;
    for (int k = 1; k < LC; ++k) {
      if (hs[k] == 0.f && hs[k + 1] == 0.f) { tm = k - 1; break; }
    }
    tmax_s = tm;
  }
  __syncthreads();

  const int tmax = tmax_s;
  const int n = threadIdx.x >> 4;   // batch within group
  const int j = threadIdx.x & 15;   // time phase (coalesced over t)
  const int b = bg * 16 + n;
  const float s1 = S[((long)b * C_ + c) * 2 + 0];
  const float s2 = S[((long)b * C_ + c) * 2 + 1];
  float* yrow = y + (long)b * T_ + c * LC;
  for (int t = j; t <= tmax; t += 16) {
    yrow[t] += hs[t + 1] * s1 + (a2 * hs[t]) * s2;
  }
}

// ---------------------------------------------------------------------------
extern "C" void kernel_launch(void* const* d_in, const int* in_sizes, int n_in,
                              void* d_out, int out_size, void* d_ws,
                              size_t ws_size, hipStream_t stream) {
  const float* x      = (const float*)d_in[0];  // [B, T, 3]
  const float* carry0 = (const float*)d_in[1];  // [B, 2]
  const float* coef   = (const float*)d_in[2];  // [5]
  float* y = (float*)d_out;                     // [B, T]
  float* S = (float*)d_ws;                      // B*C*2 floats = 512 KB

  biquad_pass1<<<(BG_ * C_) / WPB, 256, 0, stream>>>(x, coef, y);
  biquad_pass2<<<1, 256, 0, stream>>>(y, carry0, coef, S);
  biquad_pass3<<<BG_ * C_, 256, 0, stream>>>(y, S, coef);
}